// MultiLayerGATv2_65017214927351
// MI455X (gfx1250) — compile-verified
//
#include <hip/hip_runtime.h>

// GATv2 x3 layers: N=50000 nodes, E=1.6M edges (+N self loops), H=2 heads, C=32.
// f32 end-to-end; dense projections use native f32 WMMA (v_wmma_f32_16x16x4_f32).

#define NEG_SLOPE 0.2f

typedef float v2f __attribute__((ext_vector_type(2)));
typedef float v8f __attribute__((ext_vector_type(8)));

// ---- order-preserving uint encoding for float atomic max -------------------
__device__ __forceinline__ unsigned enc_ord(float f) {
  unsigned u = __float_as_uint(f);
  return (u & 0x80000000u) ? ~u : (u | 0x80000000u);
}
__device__ __forceinline__ float dec_ord(unsigned u) {
  u = (u & 0x80000000u) ? (u ^ 0x80000000u) : ~u;
  return __uint_as_float(u);
}
#define ENC_NEG_INF 0x007FFFFFu  // enc(-inf)

// ---- GEMM: C[N x 64] = X[N x K] @ W[K x 64], K multiple of 4 ----------------
// One wave -> 16-row strip x 64 cols (4 WMMA tiles), K stepped by 4.
__global__ void __launch_bounds__(256) gatv2_gemm64(const float* __restrict__ X,
                                                    const float* __restrict__ W,
                                                    float* __restrict__ C,
                                                    int N, int K) {
  const int lane        = threadIdx.x & 31;
  const int waveInBlk   = threadIdx.x >> 5;
  const int wavesPerBlk = blockDim.x >> 5;
  const int strip       = blockIdx.x * wavesPerBlk + waveInBlk;
  const int m0          = strip * 16;
  if (m0 >= N) return;                    // wave-uniform: EXEC stays all-ones
  const int half = lane >> 4;             // 0: lanes 0-15, 1: lanes 16-31
  const int l16  = lane & 15;

  v8f acc0 = {}, acc1 = {}, acc2 = {}, acc3 = {};
  const float* __restrict__ xrow = X + (size_t)(m0 + l16) * K;

  for (int k0 = 0; k0 < K; k0 += 4) {
    // A fragment (16x4 f32): lanes 0-15 -> K=k0+0/k0+1, lanes 16-31 -> k0+2/k0+3
    v2f a;
    a.x = xrow[k0 + half * 2 + 0];
    a.y = xrow[k0 + half * 2 + 1];
    // B fragments (4x16 f32): v0/v1 hold rows (k0+2*half) / (k0+2*half+1)
    const float* __restrict__ wrow = W + (size_t)(k0 + half * 2) * 64 + l16;
    v2f b0, b1, b2, b3;
    b0.x = wrow[0];   b0.y = wrow[64];
    b1.x = wrow[16];  b1.y = wrow[80];
    b2.x = wrow[32];  b2.y = wrow[96];
    b3.x = wrow[48];  b3.y = wrow[112];
    acc0 = __builtin_amdgcn_wmma_f32_16x16x4_f32(false, a, false, b0, (short)0, acc0, false, false);
    acc1 = __builtin_amdgcn_wmma_f32_16x16x4_f32(false, a, false, b1, (short)0, acc1, false, false);
    acc2 = __builtin_amdgcn_wmma_f32_16x16x4_f32(false, a, false, b2, (short)0, acc2, false, false);
    acc3 = __builtin_amdgcn_wmma_f32_16x16x4_f32(false, a, false, b3, (short)0, acc3, false, false);
  }
  // D layout: VGPR i -> row m0 + half*8 + i, col = tile*16 + l16
  float* __restrict__ crow = C + (size_t)(m0 + half * 8) * 64 + l16;
#pragma unroll
  for (int i = 0; i < 8; ++i) {
    crow[(size_t)i * 64 +  0] = acc0[i];
    crow[(size_t)i * 64 + 16] = acc1[i];
    crow[(size_t)i * 64 + 32] = acc2[i];
    crow[(size_t)i * 64 + 48] = acc3[i];
  }
}

// ---- per-layer init: accum=0, denom=0, smax=enc(-inf) -----------------------
__global__ void gatv2_init(unsigned* __restrict__ smax, float* __restrict__ denom,
                           float* __restrict__ outacc, int N) {
  long long gid = (long long)blockIdx.x * blockDim.x + threadIdx.x;
  if (gid < (long long)N * 64) outacc[gid] = 0.f;
  if (gid < (long long)N * 2) { smax[gid] = ENC_NEG_INF; denom[gid] = 0.f; }
}

// ---- pass 1: s[e,h] = att_h . leakyrelu(xl[src]+xr[dst]); atomic segment max
__global__ void gatv2_scores(const int* __restrict__ ei, int E, int ET,
                             const float* __restrict__ xl, const float* __restrict__ xr,
                             const float* __restrict__ att,
                             float* __restrict__ sbuf, unsigned* __restrict__ smax) {
  long long gid = (long long)blockIdx.x * blockDim.x + threadIdx.x;
  if (gid >= (long long)ET * 2) return;
  const int e = (int)(gid >> 1), h = (int)(gid & 1);
  int s, d;
  if (e < E) { s = ei[e]; d = ei[E + e]; } else { s = d = e - E; }
  const float4* __restrict__ pl = (const float4*)(xl + (size_t)s * 64 + h * 32);
  const float4* __restrict__ pr = (const float4*)(xr + (size_t)d * 64 + h * 32);
  const float4* __restrict__ pa = (const float4*)(att + h * 32);
  float acc = 0.f;
#pragma unroll
  for (int i = 0; i < 8; ++i) {
    float4 a = pl[i], b = pr[i], w = pa[i];
    float tx = a.x + b.x, ty = a.y + b.y, tz = a.z + b.z, tw = a.w + b.w;
    tx = tx > 0.f ? tx : tx * NEG_SLOPE;
    ty = ty > 0.f ? ty : ty * NEG_SLOPE;
    tz = tz > 0.f ? tz : tz * NEG_SLOPE;
    tw = tw > 0.f ? tw : tw * NEG_SLOPE;
    acc += tx * w.x + ty * w.y + tz * w.z + tw * w.w;
  }
  sbuf[gid] = acc;
  atomicMax(smax + (size_t)d * 2 + h, enc_ord(acc));
}

// ---- pass 2: ex = exp(s - smax[dst]); denom[dst] += ex ----------------------
__global__ void gatv2_exp(const int* __restrict__ ei, int E, int ET,
                          float* __restrict__ sbuf, const unsigned* __restrict__ smax,
                          float* __restrict__ denom) {
  long long gid = (long long)blockIdx.x * blockDim.x + threadIdx.x;
  if (gid >= (long long)ET * 2) return;
  const int e = (int)(gid >> 1), h = (int)(gid & 1);
  int d;
  if (e < E) { d = ei[E + e]; } else { d = e - E; }
  const float sm = dec_ord(smax[(size_t)d * 2 + h]);
  const float ex = __expf(sbuf[gid] - sm);
  sbuf[gid] = ex;                            // in-place: s -> exp(s - max)
  unsafeAtomicAdd(denom + (size_t)d * 2 + h, ex);
}

// ---- pass 3: out[dst,c] += (ex/denom[dst,h]) * xl[src,c]; 64 lanes per edge -
__global__ void gatv2_aggregate(const int* __restrict__ ei, int E, long long ET,
                                const float* __restrict__ xl,
                                const float* __restrict__ exbuf,
                                const float* __restrict__ denom,
                                float* __restrict__ outacc) {
  long long gid = (long long)blockIdx.x * blockDim.x + threadIdx.x;
  long long el = gid >> 6;
  if (el >= ET) return;
  const int e = (int)el;
  const int c = (int)(gid & 63);
  const int h = c >> 5;
  int s, d;
  if (e < E) { s = ei[e]; d = ei[E + e]; } else { s = d = e - (int)E; }
  const float alpha = exbuf[(size_t)e * 2 + h] / denom[(size_t)d * 2 + h];
  unsafeAtomicAdd(outacc + (size_t)d * 64 + c, alpha * xl[(size_t)s * 64 + c]);
}

// ---- epilogue: out = acc + bias, optional ReLU ------------------------------
__global__ void gatv2_bias_act(const float* __restrict__ acc, const float* __restrict__ b,
                               float* __restrict__ out, long long tot, int relu) {
  long long gid = (long long)blockIdx.x * blockDim.x + threadIdx.x;
  if (gid >= tot) return;
  float v = acc[gid] + b[gid & 63];
  if (relu) v = fmaxf(v, 0.f);
  out[gid] = v;
}

// ---------------------------------------------------------------------------
static void run_layer(const float* hin, int K, const float* Wl, const float* Wr,
                      const float* att, const float* bias, float* hout, int relu,
                      int N, int E, int ET,
                      float* xl, float* xr, float* outacc, float* exbuf,
                      float* denom, unsigned* smax, const int* ei, hipStream_t stream) {
  const int strips = (N + 15) / 16;
  dim3 gblk(256), ggrd((unsigned)((strips + 7) / 8));
  hipLaunchKernelGGL(gatv2_gemm64, ggrd, gblk, 0, stream, hin, Wl, xl, N, K);
  hipLaunchKernelGGL(gatv2_gemm64, ggrd, gblk, 0, stream, hin, Wr, xr, N, K);

  const long long n64 = (long long)N * 64;
  gatv2_init<<<(unsigned)((n64 + 255) / 256), 256, 0, stream>>>(smax, denom, outacc, N);

  const long long eh = (long long)ET * 2;
  gatv2_scores<<<(unsigned)((eh + 255) / 256), 256, 0, stream>>>(ei, E, ET, xl, xr, att, exbuf, smax);
  gatv2_exp<<<(unsigned)((eh + 255) / 256), 256, 0, stream>>>(ei, E, ET, exbuf, smax, denom);

  const long long ec = (long long)ET * 64;
  gatv2_aggregate<<<(unsigned)((ec + 255) / 256), 256, 0, stream>>>(ei, E, (long long)ET, xl, exbuf, denom, outacc);
  gatv2_bias_act<<<(unsigned)((n64 + 255) / 256), 256, 0, stream>>>(outacc, bias, hout, n64, relu);
}

extern "C" void kernel_launch(void* const* d_in, const int* in_sizes, int n_in,
                              void* d_out, int out_size, void* d_ws, size_t ws_size,
                              hipStream_t stream) {
  const float* x    = (const float*)d_in[0];
  const int*   ei   = (const int*)d_in[1];
  const float* Wl1  = (const float*)d_in[2];
  const float* Wr1  = (const float*)d_in[3];
  const float* att1 = (const float*)d_in[4];
  const float* b1   = (const float*)d_in[5];
  const float* Wl2  = (const float*)d_in[6];
  const float* Wr2  = (const float*)d_in[7];
  const float* att2 = (const float*)d_in[8];
  const float* b2   = (const float*)d_in[9];
  const float* Wl3  = (const float*)d_in[10];
  const float* Wr3  = (const float*)d_in[11];
  const float* att3 = (const float*)d_in[12];
  const float* b3   = (const float*)d_in[13];

  const int E  = in_sizes[1] / 2;
  const int K1 = in_sizes[2] / 64;   // 128
  const int N  = in_sizes[0] / K1;   // 50000
  const int ET = E + N;              // edges + self loops

  // workspace layout (f32 unless noted)
  float* xl     = (float*)d_ws;                     // N*64
  float* xr     = xl + (size_t)N * 64;              // N*64
  float* outacc = xr + (size_t)N * 64;              // N*64
  float* hbuf   = outacc + (size_t)N * 64;          // N*64
  float* exbuf  = hbuf + (size_t)N * 64;            // ET*2 (scores, then exp)
  float* denom  = exbuf + (size_t)ET * 2;           // N*2
  unsigned* smax = (unsigned*)(denom + (size_t)N * 2); // N*2 (ordered-uint max)

  run_layer(x,    K1, Wl1, Wr1, att1, b1, hbuf, 1, N, E, ET,
            xl, xr, outacc, exbuf, denom, smax, ei, stream);
  run_layer(hbuf, 64, Wl2, Wr2, att2, b2, hbuf, 1, N, E, ET,
            xl, xr, outacc, exbuf, denom, smax, ei, stream);
  run_layer(hbuf, 64, Wl3, Wr3, att3, b3, (float*)d_out, 0, N, E, ET,
            xl, xr, outacc, exbuf, denom, smax, ei, stream);
  (void)n_in; (void)out_size; (void)ws_size;
}